// Attention_27006754358036
// MI455X (gfx1250) — compile-verified
//
#include <hip/hip_runtime.h>

// CDNA5 / gfx1250, wave32. f32 WMMA path (V_WMMA_F32_16X16X4_F32).
typedef float v2f __attribute__((ext_vector_type(2)));
typedef float v8f __attribute__((ext_vector_type(8)));

namespace {
constexpr int D    = 256;
constexpr int T    = 1024;
constexpr int NK   = 1024;
constexpr int NB   = 32;
constexpr int WIN  = 100;
constexpr int WINP = 112;   // window padded to 7 tiles of 16 keys
constexpr int DP   = 260;   // padded LDS row stride (floats) for Q/K tiles
constexpr int PSTR = 116;   // padded LDS stride for score rows
constexpr long RES_ELEMS = (long)NB * T * (2 * D);  // (B,T,512)
constexpr long ALN_ELEMS = (long)NB * NK * T;       // (B,N,T)
}

// ---- kernel 1: zero the alignments region (134 MB, float4 stores) ----
__global__ __launch_bounds__(256) void zero_align_kernel(float* __restrict__ aln, long n4) {
  long i = (long)blockIdx.x * blockDim.x + threadIdx.x;
  if (i < n4) ((float4*)aln)[i] = make_float4(0.f, 0.f, 0.f, 0.f);
}

// ---- kernel 2: windowed attention, 1 block = 64 queries of one batch ----
__global__ __launch_bounds__(128) void win_attn_kernel(
    const float* __restrict__ Q, const float* __restrict__ K,
    const float* __restrict__ V, const int* __restrict__ PrevArr,
    float* __restrict__ out)
{
  __shared__ float q_lds[64 * DP];     //  66,560 B
  __shared__ float k_lds[WINP * DP];   // 116,480 B
  __shared__ float p_lds[64 * PSTR];   //  29,696 B  (~208 KB of 320 KB WGP LDS)

  const int tid  = threadIdx.x;
  const int lane = tid & 31;
  const int w    = tid >> 5;        // wave id 0..3
  const int half = lane >> 4;       // 0: lanes 0-15, 1: lanes 16-31
  const int row  = lane & 15;       // M-row (A) or N-col (B/C/D) index
  const int b    = blockIdx.x >> 4;
  const int q0   = (blockIdx.x & 15) * 64;
  const int prev = PrevArr[b];
  const int w16  = w * 16;

  const float* Qb = Q + ((long)b * T + q0) * D;
  const float* Kb = K + (long)b * NK * D;
  const float* Vb = V + (long)b * NK * D;

  // stage K window rows [prev, prev+112) (clamped; padded rows masked later)
  for (int i = tid; i < WINP * (D / 4); i += 128) {
    int r = i >> 6;                  // D/4 = 64 float4 per row
    int c = (i & 63) << 2;
    int gk = min(prev + r, NK - 1);
    float4 val = *(const float4*)(Kb + (long)gk * D + c);
    *(float4*)(k_lds + r * DP + c) = val;
  }
  // stage 64-query Q tile
  for (int i = tid; i < 64 * (D / 4); i += 128) {
    int r = i >> 6;
    int c = (i & 63) << 2;
    float4 val = *(const float4*)(Qb + (long)r * D + c);
    *(float4*)(q_lds + r * DP + c) = val;
  }
  __syncthreads();

  // ---- phase A: S = (Q · K_winᵀ) * 1/16 via V_WMMA_F32_16X16X4_F32 ----
  {
    v8f acc[7];
    #pragma unroll
    for (int nt = 0; nt < 7; ++nt) acc[nt] = v8f{0.f,0.f,0.f,0.f,0.f,0.f,0.f,0.f};

    #pragma unroll 4
    for (int kd = 0; kd < 64; ++kd) {
      const int kb = kd * 4 + half * 2;   // lane's K-pair base within 16x16x4 step
      v2f a = *(const v2f*)(q_lds + (w16 + row) * DP + kb);   // A[m][k..k+1]
      #pragma unroll
      for (int nt = 0; nt < 7; ++nt) {
        v2f bb = *(const v2f*)(k_lds + (nt * 16 + row) * DP + kb);  // B[k][n]=K[n][k]
        acc[nt] = __builtin_amdgcn_wmma_f32_16x16x4_f32(
            false, a, false, bb, (short)0, acc[nt], false, false);
      }
    }
    // scatter scaled scores to LDS: C/D layout -> M = r + 8*half, N = nt*16 + row
    #pragma unroll
    for (int nt = 0; nt < 7; ++nt) {
      #pragma unroll
      for (int r = 0; r < 8; ++r)
        p_lds[(w16 + r + half * 8) * PSTR + nt * 16 + row] = acc[nt][r] * 0.0625f;
    }
  }

  // ---- softmax + argmax per query row (2 lanes/row, 56 cols each) ----
  float* srow = p_lds + (w16 + row) * PSTR;
  const int j0 = half * 56;
  float M = -3.0e38f;
  int   arg = 0;
  for (int jj = 0; jj < 56; ++jj) {
    int j = j0 + jj;
    if (j < WIN) {
      float s = srow[j];
      if (s > M) { M = s; arg = j; }    // strict > keeps first occurrence
    }
  }
  {
    float oM = __shfl_xor(M, 16, 32);
    int   oA = __shfl_xor(arg, 16, 32);
    if (oM > M || (oM == M && oA < arg)) { M = oM; arg = oA; }
  }
  float sum = 0.f;
  for (int jj = 0; jj < 56; ++jj) {
    int j = j0 + jj;
    float p = 0.f;
    if (j < WIN) p = __expf(srow[j] - M);
    srow[j] = p;                       // padded cols j>=100 become exactly 0
    sum += p;
  }
  sum += __shfl_xor(sum, 16, 32);
  const float inv = 1.0f / sum;
  for (int jj = 0; jj < 56; ++jj) srow[j0 + jj] *= inv;

  if (half == 0)
    out[RES_ELEMS + ALN_ELEMS + (long)b * T + q0 + w16 + row] = (float)(prev + arg);

  __syncthreads();

  // ---- coalesced scatter of window probs into alignments[b, prev+j, t] ----
  for (int i = tid; i < WIN * 64; i += 128) {
    int j = i >> 6;
    int toff = i & 63;   // query-in-block == p_lds row index
    out[RES_ELEMS + ((long)b * NK + prev + j) * T + q0 + toff] =
        p_lds[toff * PSTR + j];
  }

  // ---- phase B: result[:, :256] = P · V_win via f32 WMMA ----
  for (int dt = 0; dt < 16; ++dt) {
    v8f acc = v8f{0.f,0.f,0.f,0.f,0.f,0.f,0.f,0.f};
    const float* vcol = Vb + dt * 16 + row;   // B-frag N-col = dim, rows = keys
    #pragma unroll 4
    for (int kc = 0; kc < 28; ++kc) {
      v2f a = *(const v2f*)(srow + kc * 4 + half * 2);   // A = P[m][k..k+1]
      int k0 = prev + kc * 4 + half * 2;                 // global key row
      v2f bb;
      bb.x = vcol[(long)min(k0,     NK - 1) * D];        // p==0 past window,
      bb.y = vcol[(long)min(k0 + 1, NK - 1) * D];        // clamp avoids OOB
      acc = __builtin_amdgcn_wmma_f32_16x16x4_f32(
          false, a, false, bb, (short)0, acc, false, false);
    }
    #pragma unroll
    for (int r = 0; r < 8; ++r) {
      int t = q0 + w16 + r + half * 8;
      out[((long)b * T + t) * (2 * D) + dt * 16 + row] = acc[r];
    }
  }

  // ---- result[:, 256:512] = Q (copy from LDS, float4) ----
  for (int i = tid; i < 64 * (D / 4); i += 128) {
    int r = i >> 6;
    int c = (i & 63) << 2;
    float4 val = *(const float4*)(q_lds + r * DP + c);
    *(float4*)(out + ((long)b * T + q0 + r) * (2 * D) + D + c) = val;
  }
}

extern "C" void kernel_launch(void* const* d_in, const int* in_sizes, int n_in,
                              void* d_out, int out_size, void* d_ws, size_t ws_size,
                              hipStream_t stream) {
  const float* q    = (const float*)d_in[0];
  const float* k    = (const float*)d_in[1];
  const float* v    = (const float*)d_in[2];
  const int*   prev = (const int*)d_in[3];
  float* out = (float*)d_out;

  const long n4 = ALN_ELEMS / 4;
  zero_align_kernel<<<dim3((unsigned)((n4 + 255) / 256)), dim3(256), 0, stream>>>(
      out + RES_ELEMS, n4);
  win_attn_kernel<<<dim3(NB * (T / 64)), dim3(128), 0, stream>>>(q, k, v, prev, out);
}